// DiffusionEncoder_8340826488877
// MI455X (gfx1250) — compile-verified
//
#include <hip/hip_runtime.h>
#include <hip/hip_bf16.h>
#include <cstdint>
#include <cstddef>

#define N_NODES   10000
#define N_EDGES   640000
#define NODE_DIM  64
#define HIDDEN    128
#define NUM_LAYERS 4
#define E_TOTAL   (N_EDGES + N_NODES)   // 650000, multiple of 16

typedef float v2f __attribute__((ext_vector_type(2)));
typedef float v8f __attribute__((ext_vector_type(8)));

// D = A(16x4 f32) * B(4x16 f32) + C(16x16 f32), full fp32 — exact vs reference
__device__ __forceinline__ v8f wmma_k4(v2f a, v2f b, v8f c) {
    return __builtin_amdgcn_wmma_f32_16x16x4_f32(
        false, a, false, b, (short)0, c, false, false);
}

// ---------------------------------------------------------------------------
// Repack W[K,128] (row-major) into fragment-native pairs:
//   Wp[(k>>1)*128 + col][j] = W[(k&~1)+j][col],  j = k&1
// so one B fragment (kb, kb+1) x 16 cols is a contiguous, aligned b64 per lane.
// ---------------------------------------------------------------------------
__global__ void repack_w_kernel(const float* __restrict__ W,
                                float* __restrict__ Wp, int K) {
    int i = blockIdx.x * blockDim.x + threadIdx.x;
    if (i >= K * HIDDEN) return;
    int k = i / HIDDEN;
    int c = i - k * HIDDEN;
    Wp[((size_t)(k >> 1) * HIDDEN + c) * 2 + (k & 1)] = W[i];
}

// ---------------------------------------------------------------------------
// Fill buffer with -FLT_MAX (segment-max identity; self loops cover all nodes)
// ---------------------------------------------------------------------------
__global__ void fill_negmax_kernel(float* __restrict__ p, int n) {
    int i = blockIdx.x * blockDim.x + threadIdx.x;
    if (i < n) p[i] = -3.402823466e38f;
}

// ---------------------------------------------------------------------------
// out[M,128] = A[M,K] @ W + bias, W pre-packed. One wave per 16-row tile.
// ---------------------------------------------------------------------------
__global__ void linear128_wmma_kernel(const float* __restrict__ A,
                                      const float* __restrict__ Wp,
                                      const float* __restrict__ bias,
                                      float* __restrict__ out,
                                      int M, int K) {
    const int lane = threadIdx.x & 31;
    const int wave = threadIdx.x >> 5;
    const int tile = blockIdx.x * 4 + wave;
    if (tile * 16 >= M) return;

    const int row = lane & 15;
    const int grp = lane >> 4;
    const float* arow = A + (size_t)(tile * 16 + row) * K;
    const v2f* Wv = (const v2f*)Wp;     // indexed by p*128 + col

    v8f c[8];
#pragma unroll
    for (int nt = 0; nt < 8; ++nt) {
        float bv = bias[nt * 16 + row];
#pragma unroll
        for (int r = 0; r < 8; ++r) c[nt][r] = bv;
    }

#pragma unroll 2
    for (int k0 = 0; k0 < K; k0 += 4) {
        const int kb = k0 + 2 * grp;
        const int p  = kb >> 1;
        v2f a = *(const v2f*)(arow + kb);
        v2f b[8];
#pragma unroll
        for (int nt = 0; nt < 8; ++nt)
            b[nt] = Wv[(size_t)p * HIDDEN + nt * 16 + row];
#pragma unroll
        for (int nt = 0; nt < 8; ++nt)
            c[nt] = wmma_k4(a, b[nt], c[nt]);
    }

#pragma unroll
    for (int nt = 0; nt < 8; ++nt)
#pragma unroll
        for (int r = 0; r < 8; ++r)
            out[(size_t)(tile * 16 + r + 8 * grp) * HIDDEN + nt * 16 + row] = c[nt][r];
}

// ---------------------------------------------------------------------------
// EdgeConv layer, one wave per 16-edge tile:
//   m = relu(concat[x_i, x_j - x_i] @ W1 + b1) @ W2 + b2
//   hout[dst] = atomic-max(m)
// ---------------------------------------------------------------------------
__global__ void edge_conv_kernel(const float* __restrict__ h,
                                 const long long* __restrict__ esrc,
                                 const long long* __restrict__ edst,
                                 const float* __restrict__ Wp1,
                                 const float* __restrict__ b1,
                                 const float* __restrict__ Wp2,
                                 const float* __restrict__ b2,
                                 float* __restrict__ hout) {
    __shared__ float m1s[4][16 * HIDDEN];       // 32 KB: GEMM1->GEMM2 re-fragment

    const int lane = threadIdx.x & 31;
    const int wave = threadIdx.x >> 5;
    const int tile = blockIdx.x * 4 + wave;
    if (tile * 16 >= E_TOTAL) return;           // whole wave exits together

    const int row = lane & 15;
    const int grp = lane >> 4;

    int e = tile * 16 + row;
    if (e >= E_TOTAL) e = E_TOTAL - 1;          // harmless dup under max-agg
    int s, d;
    if (e < N_EDGES) { s = (int)esrc[e]; d = (int)edst[e]; }
    else             { s = d = e - N_EDGES; }

    const float* hd = h + (size_t)d * HIDDEN;
    const float* hs = h + (size_t)s * HIDDEN;
    const v2f* W1v = (const v2f*)Wp1;
    const v2f* W2v = (const v2f*)Wp2;

    // ---- GEMM1: [16 x 256] @ W1[256 x 128] + b1 -------------------------
    v8f c[8];
#pragma unroll
    for (int nt = 0; nt < 8; ++nt) {
        float bv = b1[nt * 16 + row];
#pragma unroll
        for (int r = 0; r < 8; ++r) c[nt][r] = bv;
    }

    // first half of K: columns are x_i
#pragma unroll 2
    for (int k0 = 0; k0 < HIDDEN; k0 += 4) {
        const int kb = k0 + 2 * grp;
        const int p  = kb >> 1;
        v2f a = *(const v2f*)(hd + kb);
        v2f b[8];
#pragma unroll
        for (int nt = 0; nt < 8; ++nt)
            b[nt] = W1v[(size_t)p * HIDDEN + nt * 16 + row];
#pragma unroll
        for (int nt = 0; nt < 8; ++nt)
            c[nt] = wmma_k4(a, b[nt], c[nt]);
    }

    // second half of K: columns are x_j - x_i
#pragma unroll 2
    for (int k0 = HIDDEN; k0 < 2 * HIDDEN; k0 += 4) {
        const int kb = k0 + 2 * grp;
        const int p  = kb >> 1;
        v2f aj = *(const v2f*)(hs + (kb - HIDDEN));
        v2f ai = *(const v2f*)(hd + (kb - HIDDEN));
        v2f a;
        a.x = aj.x - ai.x;
        a.y = aj.y - ai.y;
        v2f b[8];
#pragma unroll
        for (int nt = 0; nt < 8; ++nt)
            b[nt] = W1v[(size_t)p * HIDDEN + nt * 16 + row];
#pragma unroll
        for (int nt = 0; nt < 8; ++nt)
            c[nt] = wmma_k4(a, b[nt], c[nt]);
    }

    // ---- ReLU, stage through this wave's private LDS slice --------------
    float* ms = m1s[wave];
#pragma unroll
    for (int nt = 0; nt < 8; ++nt)
#pragma unroll
        for (int r = 0; r < 8; ++r) {
            float v = c[nt][r];
            v = v > 0.0f ? v : 0.0f;
            ms[(r + 8 * grp) * HIDDEN + nt * 16 + row] = v;
        }
    // wave-private slice: only need this wave's own DS stores complete
    asm volatile("s_wait_dscnt 0" ::: "memory");

    // ---- GEMM2: [16 x 128] @ W2[128 x 128] + b2 -------------------------
#pragma unroll
    for (int nt = 0; nt < 8; ++nt) {
        float bv = b2[nt * 16 + row];
#pragma unroll
        for (int r = 0; r < 8; ++r) c[nt][r] = bv;
    }

#pragma unroll 2
    for (int k0 = 0; k0 < HIDDEN; k0 += 4) {
        const int kb = k0 + 2 * grp;
        const int p  = kb >> 1;
        v2f a = *(const v2f*)(ms + row * HIDDEN + kb);
        v2f b[8];
#pragma unroll
        for (int nt = 0; nt < 8; ++nt)
            b[nt] = W2v[(size_t)p * HIDDEN + nt * 16 + row];
#pragma unroll
        for (int nt = 0; nt < 8; ++nt)
            c[nt] = wmma_k4(a, b[nt], c[nt]);
    }

    // ---- scatter-max into hout[dst] (idempotent under duplicates) -------
#pragma unroll
    for (int r = 0; r < 8; ++r) {
        const int m = r + 8 * grp;              // C-layout row of this element
        const int dnode = __shfl(d, m, 32);     // dst of that edge row
        float* orow = hout + (size_t)dnode * HIDDEN;
#pragma unroll
        for (int nt = 0; nt < 8; ++nt)
            atomicMax(orow + nt * 16 + row, c[nt][r]);
    }
}

// ---------------------------------------------------------------------------
extern "C" void kernel_launch(void* const* d_in, const int* in_sizes, int n_in,
                              void* d_out, int out_size, void* d_ws, size_t ws_size,
                              hipStream_t stream) {
    const float*     x     = (const float*)d_in[0];
    const long long* eidx  = (const long long*)d_in[1];   // int64 [2, N_EDGES]
    const float*     W_emb = (const float*)d_in[2];
    const float*     b_emb = (const float*)d_in[3];
    const float*     W1    = (const float*)d_in[4];       // [4, 256, 128]
    const float*     b1    = (const float*)d_in[5];       // [4, 128]
    const float*     W2    = (const float*)d_in[6];       // [4, 128, 128]
    const float*     b2    = (const float*)d_in[7];       // [4, 128]
    const float*     W_fc  = (const float*)d_in[8];
    const float*     b_fc  = (const float*)d_in[9];
    float*           out   = (float*)d_out;

    const long long* esrc = eidx;
    const long long* edst = eidx + N_EDGES;

    // workspace layout
    const size_t NH = (size_t)N_NODES * HIDDEN;
    float* hA     = (float*)d_ws;
    float* hB     = hA + NH;
    float* Wp_emb = hB + NH;                                  // 64*128
    float* Wp1    = Wp_emb + (size_t)NODE_DIM * HIDDEN;       // 4 * 256*128
    float* Wp2    = Wp1 + (size_t)NUM_LAYERS * 2 * HIDDEN * HIDDEN; // 4 * 128*128
    float* Wp_fc  = Wp2 + (size_t)NUM_LAYERS * HIDDEN * HIDDEN;     // 128*128

    const dim3 blk(128);                               // 4 waves / block
    const int node_tiles = N_NODES / 16;               // 625
    const int edge_tiles = E_TOTAL / 16;               // 40625
    const int node_elems = N_NODES * HIDDEN;

    // ---- repack all weights into fragment-native pair layout ------------
    {
        int nE = NODE_DIM * HIDDEN;
        repack_w_kernel<<<(nE + 255) / 256, 256, 0, stream>>>(W_emb, Wp_emb, NODE_DIM);
        int n1 = 2 * HIDDEN * HIDDEN;
        int n2 = HIDDEN * HIDDEN;
        for (int l = 0; l < NUM_LAYERS; ++l) {
            repack_w_kernel<<<(n1 + 255) / 256, 256, 0, stream>>>(
                W1 + (size_t)l * n1, Wp1 + (size_t)l * n1, 2 * HIDDEN);
            repack_w_kernel<<<(n2 + 255) / 256, 256, 0, stream>>>(
                W2 + (size_t)l * n2, Wp2 + (size_t)l * n2, HIDDEN);
        }
        repack_w_kernel<<<(n2 + 255) / 256, 256, 0, stream>>>(W_fc, Wp_fc, HIDDEN);
    }

    // h0 = x @ W_emb + b_emb
    linear128_wmma_kernel<<<(node_tiles + 3) / 4, blk, 0, stream>>>(
        x, Wp_emb, b_emb, hA, N_NODES, NODE_DIM);

    float* cur = hA;
    float* nxt = hB;
    for (int l = 0; l < NUM_LAYERS; ++l) {
        fill_negmax_kernel<<<(node_elems + 255) / 256, 256, 0, stream>>>(nxt, node_elems);
        edge_conv_kernel<<<(edge_tiles + 3) / 4, blk, 0, stream>>>(
            cur, esrc, edst,
            Wp1 + (size_t)l * 2 * HIDDEN * HIDDEN, b1 + (size_t)l * HIDDEN,
            Wp2 + (size_t)l * HIDDEN * HIDDEN,     b2 + (size_t)l * HIDDEN,
            nxt);
        float* t = cur; cur = nxt; nxt = t;
    }

    // out = h @ W_fc + b_fc
    linear128_wmma_kernel<<<(node_tiles + 3) / 4, blk, 0, stream>>>(
        cur, Wp_fc, b_fc, out, N_NODES, HIDDEN);
}